// PoolCliqueToLine_16217796509771
// MI455X (gfx1250) — compile-verified
//
#include <hip/hip_runtime.h>
#include <hip/hip_bf16.h>

#define N_IN     16384
#define N_OUT    32768
#define N_EDGES  131072
#define BATCH    4
#define FEATS    128
#define BF       (BATCH * FEATS)              // 512 feature slabs
#define THREADS  256
#define COL_CHUNKS 4                          // col-range split for occupancy
#define COLS_PER_CHUNK (N_OUT / COL_CHUNKS)   // 8192

typedef int v4i __attribute__((ext_vector_type(4)));
#define AS_GLOBAL __attribute__((address_space(1)))
#define AS_LDS    __attribute__((address_space(3)))

// lower_bound over sorted int array
__device__ __forceinline__ int lower_bound_dev(const int* __restrict__ a,
                                               int n, int key) {
    int lo = 0;
    while (n > 0) {
        int half = n >> 1;
        int mid  = lo + half;
        if (a[mid] < key) { lo = mid + 1; n -= half + 1; }
        else               { n = half; }
    }
    return lo;
}

// ---------------------------------------------------------------------------
// Kernel A: segment starts. col_start[c] = first edge index with cols[e] >= c,
// for c in [0, N_OUT] inclusive (N_OUT+1 entries). Trivial cost (33K threads).
// ---------------------------------------------------------------------------
__global__ void PoolColStart_kernel(const int* __restrict__ cols,
                                    int* __restrict__ col_start) {
    int c = blockIdx.x * blockDim.x + threadIdx.x;
    if (c <= N_OUT) col_start[c] = lower_bound_dev(cols, N_EDGES, c);
}

// ---------------------------------------------------------------------------
// Kernel B: block = (bf slab, col chunk). Stage x[bf,:] (64 KB) into LDS with
// gfx1250 async global->LDS loads, then each thread reduces whole column
// segments: out[bf][c] = max(0, max_j sx[rows[j]]), written exactly once.
// No atomics, no zero-init, coalesced stores.
// ---------------------------------------------------------------------------
__global__ __launch_bounds__(THREADS)
void PoolCliqueToLine_kernel(const float* __restrict__ x,
                             const int*  __restrict__ rows,
                             const int*  __restrict__ cols,
                             const int*  __restrict__ col_start,
                             float*      __restrict__ out,
                             int use_ws) {
    __shared__ float sx[N_IN];                // 64 KB of the 320 KB/WGP LDS

    const int bf  = blockIdx.x;               // 0..511
    const int c0  = blockIdx.y * COLS_PER_CHUNK;
    const int tid = threadIdx.x;

    const float* xs = x + (size_t)bf * N_IN;

    // ---- Stage x[bf, :] into LDS (gfx1250 async global->LDS, ASYNCcnt) ----
#if __has_builtin(__builtin_amdgcn_global_load_async_to_lds_b128)
    for (int i = tid * 4; i < N_IN; i += THREADS * 4) {
        __builtin_amdgcn_global_load_async_to_lds_b128(
            (AS_GLOBAL v4i*)(xs + i),
            (AS_LDS    v4i*)&sx[i],
            /*offset=*/0, /*cpol=*/0);
    }
#if __has_builtin(__builtin_amdgcn_s_wait_asynccnt)
    __builtin_amdgcn_s_wait_asynccnt(0);
#else
    asm volatile("s_wait_asynccnt 0" ::: "memory");
#endif
#else
    for (int i = tid * 4; i < N_IN; i += THREADS * 4) {
        const float4 t = *(const float4*)(xs + i);
        *(float4*)&sx[i] = t;
    }
#endif
    __syncthreads();

    float* out_bf = out + (size_t)bf * N_OUT;

    // COLS_PER_CHUNK % THREADS == 0 -> no tail divergence on the outer loop.
    for (int c = c0 + tid; c < c0 + COLS_PER_CHUNK; c += THREADS) {
        int s, e;
        if (use_ws) {                 // coalesced segment-pointer loads
            s = col_start[c];
            e = col_start[c + 1];
        } else {                      // fallback: search in-kernel
            s = lower_bound_dev(cols, N_EDGES, c);
            e = lower_bound_dev(cols, N_EDGES, c + 1);
        }

        // Reference init is 0, so max starts at 0 (also covers empty segment).
        float m = 0.0f;
        for (int j = s; j < e; ++j) {
            m = fmaxf(m, sx[rows[j]]);   // rows[j]: warp-contiguous loads;
        }                                // sx[.]  : 1 random LDS gather/edge
        out_bf[c] = m;                   // each output element written once
    }
}

extern "C" void kernel_launch(void* const* d_in, const int* in_sizes, int n_in,
                              void* d_out, int out_size, void* d_ws, size_t ws_size,
                              hipStream_t stream) {
    const float* x    = (const float*)d_in[0];
    const int*   rows = (const int*)d_in[1];
    const int*   cols = (const int*)d_in[2];
    float*       out  = (float*)d_out;

    const size_t need_ws = (size_t)(N_OUT + 1) * sizeof(int);
    const int use_ws = (d_ws != nullptr && ws_size >= need_ws) ? 1 : 0;
    int* col_start = (int*)d_ws;

    if (use_ws) {
        const int n = N_OUT + 1;
        PoolColStart_kernel<<<(n + THREADS - 1) / THREADS, THREADS, 0, stream>>>(
            cols, col_start);
    }

    dim3 grid(BF, COL_CHUNKS);
    PoolCliqueToLine_kernel<<<grid, THREADS, 0, stream>>>(
        x, rows, cols, col_start, out, use_ws);
}